// ExpandedResolventFMNet_36395552866817
// MI455X (gfx1250) — compile-verified
//
#include <hip/hip_runtime.h>
#include <hip/hip_bf16.h>

// ---------------------------------------------------------------------------
// ExpandedResolventFMNet — Kronecker-structured functional-map solve, CDNA5.
//
// Math (see round-0 derivation):
//   A   = Px @ fx                      [64,128]
//   PyF = Py @ fy                      [64,128]
//   Q   = My^T My, AA = A A^T          [64,64]
//   RHS = Q @ PyF @ A^T                [64,64]   (= My^T (My PyF) A^T)
//   op  = kron(S0,Q) + L*kron(N,T) - L*kron(Pr,Ur) - L*kron(Pi,Ui)   (SPD)
//   solve op vec(X) = vec(RHS) by CG; matvec via kron(X,Y)vec(V)=vec(Y V X).
//   Output Cxy = X (row-major).
// ---------------------------------------------------------------------------

typedef __attribute__((ext_vector_type(2))) float v2f;
typedef __attribute__((ext_vector_type(4))) float v4f;
typedef __attribute__((ext_vector_type(8))) float v8f;
typedef __attribute__((ext_vector_type(4))) int   v4i;

#define K_DIM 64
#define C_DIM 128
#define V_DIM 5000
#define LMBDA 100.0f
#define CG_ITERS 320
#define KSTEP 8          // k-rows staged per LDS panel in K1
#define SPLITS 5         // split-K factor for K1 (5000/5 = 1000, 1000%8==0)
// Padded LDS row stride for 64x64 WMMA operand matrices in k_cg:
//  A-reads:  (m*12 + k) mod 64  -> 16 distinct banks (conflict-free)
//  B-reads:  upper half-wave shifted 2*76 mod 64 = 24 banks (disjoint)
//  76 % 4 == 0 keeps rows 16B-aligned for b128 copies.
#define LDSTR 76

// ---------------------------------------------------------------------------
// CDNA5 async global->LDS copy, probe-verified on this toolchain:
// global_load_async_to_lds_b128 (v4i AS(1) src, v4i AS(3) dst, imm, cpol).
// ---------------------------------------------------------------------------
#if defined(__gfx1250__) &&                                        \
    __has_builtin(__builtin_amdgcn_global_load_async_to_lds_b128) && \
    __has_builtin(__builtin_amdgcn_s_wait_asynccnt)
#define USE_ASYNC_LDS 1
#else
#define USE_ASYNC_LDS 0
#endif

#if USE_ASYNC_LDS
typedef __attribute__((address_space(1))) v4i glb_v4i;
typedef __attribute__((address_space(3))) v4i lds_v4i;
__device__ __forceinline__ void cp16_to_lds(void* lds_dst, const void* gsrc) {
  __builtin_amdgcn_global_load_async_to_lds_b128(
      (glb_v4i*)gsrc, (lds_v4i*)lds_dst, 0, 0);
}
__device__ __forceinline__ void cp_wait() {
  __builtin_amdgcn_s_wait_asynccnt(0);
}
#else
__device__ __forceinline__ void cp16_to_lds(void* lds_dst, const void* gsrc) {
  *(v4f*)lds_dst = *(const v4f*)gsrc;
}
__device__ __forceinline__ void cp_wait() {}
#endif

__device__ __forceinline__ v8f wmma4(v2f a, v2f b, v8f c) {
  // D = A(16x4,f32) * B(4x16,f32) + C(16x16,f32)
  return __builtin_amdgcn_wmma_f32_16x16x4_f32(
      /*neg_a=*/false, a, /*neg_b=*/false, b,
      /*c_mod=*/(short)0, c, /*reuse_a=*/false, /*reuse_b=*/false);
}

// ---------------------------------------------------------------------------
// K1: split-K spectral GEMM.  pbuf[(mat*splits+s)*8192 + i] = partial of
// (mat==0 ? Px@fx : Py@fy) over k-chunk s.  grid = 8*splits blocks x 256.
// Panels of KSTEP k-rows are double-buffered in LDS via async copies; each of
// the 8 waves owns one 16x16 output tile (col tile = wave id).
// ---------------------------------------------------------------------------
__global__ void k_spectral_gemm(const float* __restrict__ fx,
                                const float* __restrict__ fy,
                                const float* __restrict__ Px,
                                const float* __restrict__ Py,
                                float* __restrict__ pbuf, int splits) {
  __shared__ __attribute__((aligned(16))) float As[2][16][KSTEP];
  __shared__ __attribute__((aligned(16))) float Fs[2][KSTEP][C_DIM];

  const int b    = blockIdx.x;
  const int s    = b >> 3;                 // split index
  const int sub  = b & 7;
  const int mat  = (sub >= 4) ? 1 : 0;
  const float* P = mat ? Py : Px;
  const float* F = mat ? fy : fx;
  const int rt   = sub & 3;                // 16-row tile of P
  const int tid  = threadIdx.x;            // 256
  const int wave = tid >> 5;               // col tile (16 of 128)
  const int lane = tid & 31;
  const int n    = lane & 15;              // A row / B col within tile
  const int klo  = (lane >> 4) << 1;       // per-lane k sub-offset {0,2}

  const int kchunk = V_DIM / splits;       // 1000 or 5000 (both %KSTEP==0)
  const int kbase0 = s * kchunk;
  const int steps  = kchunk / KSTEP;

  auto load_panel = [&](int pb, int st) {
    const int kb = kbase0 + st * KSTEP;
    {  // F panel: KSTEP x 128 floats; one b128 per thread
      const int row = tid >> 5;            // 0..7
      const int c4  = (lane) << 2;         // 0..124
      cp16_to_lds(&Fs[pb][row][c4], F + (size_t)(kb + row) * C_DIM + c4);
    }
    if (tid < 32) {                        // A panel: 16 x KSTEP floats
      const int m2 = tid >> 1;
      const int k4 = (tid & 1) << 2;
      cp16_to_lds(&As[pb][m2][k4], P + (size_t)(rt * 16 + m2) * V_DIM + kb + k4);
    }
  };

  v8f acc = {};
  load_panel(0, 0);
  cp_wait();
  __syncthreads();

  int buf = 0;
  for (int st = 0; st < steps; ++st) {
    if (st + 1 < steps) load_panel(buf ^ 1, st + 1);
#pragma unroll
    for (int kk = 0; kk < KSTEP; kk += 4) {
      v2f a, bf;
      a.x  = As[buf][n][kk + klo];
      a.y  = As[buf][n][kk + klo + 1];
      bf.x = Fs[buf][kk + klo][wave * 16 + n];
      bf.y = Fs[buf][kk + klo + 1][wave * 16 + n];
      acc = wmma4(a, bf, acc);
    }
    cp_wait();
    __syncthreads();
    buf ^= 1;
  }

  const int rbase = rt * 16 + ((lane >> 4) << 3);
  const int c     = wave * 16 + n;
  float* outp = pbuf + (size_t)(mat * splits + s) * (K_DIM * C_DIM);
#pragma unroll
  for (int v = 0; v < 8; ++v)
    outp[(rbase + v) * C_DIM + c] = acc[v];
}

// Deterministic ordered reduction of split-K partials (no FP atomics).
__global__ void k_reduce_partials(const float* __restrict__ pbuf,
                                  float* __restrict__ A,
                                  float* __restrict__ PyF, int splits) {
  const int idx = blockIdx.x * 256 + threadIdx.x;   // 0..16383
  const int mat = idx >> 13;
  const int off = idx & 8191;
  float s = 0.f;
  for (int sp = 0; sp < splits; ++sp)
    s += pbuf[(size_t)(mat * splits + sp) * (K_DIM * C_DIM) + off];
  (mat ? PyF : A)[off] = s;
}

// ---------------------------------------------------------------------------
// K2: Q = My^T My, AA = A A^T, RHS = Q @ PyF @ A^T. One block, one-shot.
// ---------------------------------------------------------------------------
__global__ void k_smallgemms(const float* __restrict__ My,
                             const float* __restrict__ A,
                             const float* __restrict__ PyF,
                             float* __restrict__ Q,
                             float* __restrict__ AAm,
                             float* __restrict__ RHS) {
  __shared__ float Qs[K_DIM * K_DIM];
  __shared__ float W2[K_DIM * C_DIM];
  const int tid = threadIdx.x;

  for (int idx = tid; idx < K_DIM * K_DIM; idx += 256) {
    const int i = idx >> 6, j = idx & 63;
    float sq = 0.f, sa = 0.f;
    for (int k = 0; k < K_DIM; ++k)
      sq += My[k * K_DIM + i] * My[k * K_DIM + j];
    for (int c = 0; c < C_DIM; ++c)
      sa += A[i * C_DIM + c] * A[j * C_DIM + c];
    Qs[idx]  = sq;
    Q[idx]   = sq;
    AAm[idx] = sa;
  }
  __syncthreads();
  for (int idx = tid; idx < K_DIM * C_DIM; idx += 256) {   // W2 = Q @ PyF
    const int i = idx >> 7, c = idx & 127;
    float s = 0.f;
    for (int k = 0; k < K_DIM; ++k)
      s += Qs[i * K_DIM + k] * PyF[k * C_DIM + c];
    W2[idx] = s;
  }
  __syncthreads();
  for (int idx = tid; idx < K_DIM * K_DIM; idx += 256) {   // RHS = W2 @ A^T
    const int i = idx >> 6, j = idx & 63;
    float s = 0.f;
    for (int c = 0; c < C_DIM; ++c)
      s += W2[i * C_DIM + c] * A[j * C_DIM + c];
    RHS[idx] = s;
  }
}

// ---------------------------------------------------------------------------
// K3: invMx = inv(Mx) via Gauss-Jordan on [Mx | I] in LDS (Mx ~ I, no pivot).
// ---------------------------------------------------------------------------
__global__ void k_inverse(const float* __restrict__ Mx,
                          float* __restrict__ invMx) {
  __shared__ float Aug[K_DIM][2 * K_DIM + 1];
  __shared__ float fac[K_DIM];
  const int tid = threadIdx.x;  // 256

  for (int idx = tid; idx < K_DIM * K_DIM; idx += 256) {
    const int i = idx >> 6, j = idx & 63;
    Aug[i][j]         = Mx[idx];
    Aug[i][K_DIM + j] = (i == j) ? 1.f : 0.f;
  }
  __syncthreads();
  for (int p = 0; p < K_DIM; ++p) {
    const float piv = Aug[p][p];
    __syncthreads();
    if (tid < 2 * K_DIM) Aug[p][tid] = Aug[p][tid] / piv;
    __syncthreads();
    if (tid < K_DIM) fac[tid] = (tid == p) ? 0.f : Aug[tid][p];
    __syncthreads();
    for (int idx = tid; idx < K_DIM * 2 * K_DIM; idx += 256) {
      const int i = idx >> 7, j = idx & 127;
      if (i != p) Aug[i][j] -= fac[i] * Aug[p][j];
    }
    __syncthreads();
  }
  for (int idx = tid; idx < K_DIM * K_DIM; idx += 256) {
    const int i = idx >> 6, j = idx & 63;
    invMx[idx] = Aug[i][K_DIM + j];
  }
}

// ---------------------------------------------------------------------------
// K4: Kronecker factor matrices (all 64x64, symmetric).
// ---------------------------------------------------------------------------
__global__ void k_factors(const float* __restrict__ ex,
                          const float* __restrict__ ey,
                          const float* __restrict__ invMx,
                          const float* __restrict__ Q,
                          const float* __restrict__ AAm,
                          float* __restrict__ N,  float* __restrict__ Pr,
                          float* __restrict__ Pi, float* __restrict__ S0,
                          float* __restrict__ T,  float* __restrict__ Ur,
                          float* __restrict__ Ui) {
  __shared__ float r1r[K_DIM], r1i[K_DIM], r2r[K_DIM], r2i[K_DIM];
  __shared__ float smax;
  const int tid = threadIdx.x;  // 256

  if (tid == 0) {
    float s = 0.f;
    for (int i = 0; i < K_DIM; ++i) {
      s = fmaxf(s, ex[i]);
      s = fmaxf(s, ey[i]);
    }
    smax = s;
  }
  __syncthreads();
  if (tid < K_DIM) {
    const float g1 = sqrtf(ex[tid] / smax);   // GAMMA = 0.5
    const float g2 = sqrtf(ey[tid] / smax);
    r1r[tid] = g1 / (g1 * g1 + 1.f);
    r1i[tid] = 1.f / (g1 * g1 + 1.f);
    r2r[tid] = g2 / (g2 * g2 + 1.f);
    r2i[tid] = 1.f / (g2 * g2 + 1.f);
  }
  __syncthreads();

  for (int idx = tid; idx < K_DIM * K_DIM; idx += 256) {
    const int i = idx >> 6, j = idx & 63;
    float sn = 0.f, sr = 0.f, si = 0.f, sm = 0.f;
    for (int k = 0; k < K_DIM; ++k) {
      const float ab = invMx[k * K_DIM + i] * invMx[k * K_DIM + j];
      sn += ab;
      sr += ab * r1r[k];
      si += ab * r1i[k];
      sm += ab * (r1r[k] * r1r[k] + r1i[k] * r1i[k]);
    }
    N[idx]  = sn;
    Pr[idx] = sr;
    Pi[idx] = si;
    S0[idx] = AAm[idx] + LMBDA * sm;
    const float q = Q[idx];
    T[idx]  = q * (r2r[i] * r2r[j] + r2i[i] * r2i[j]);
    Ur[idx] = q * (r2r[i] + r2r[j]);
    Ui[idx] = q * (r2i[i] + r2i[j]);
  }
}

// ---------------------------------------------------------------------------
// 16x16 tile of A@B for 64x64 operands with per-matrix row strides, as one
// chain of 16 V_WMMA_F32_16X16X4_F32 ops.
// ---------------------------------------------------------------------------
__device__ __forceinline__ v8f tile_mm(const float* __restrict__ A, int sa,
                                       const float* __restrict__ B, int sb,
                                       int ti, int tj, int lane) {
  const int m  = lane & 15;
  const int kb = (lane >> 4) << 1;
  const float* Arow = A + (ti * 16 + m) * sa;
  const float* Bcol = B + tj * 16 + m;
  v8f acc = {};
#pragma unroll
  for (int k0 = 0; k0 < K_DIM; k0 += 4) {
    v2f a, b;
    a.x = Arow[k0 + kb];
    a.y = Arow[k0 + kb + 1];
    b.x = Bcol[(k0 + kb) * sb];
    b.y = Bcol[(k0 + kb + 1) * sb];
    acc = wmma4(a, b, acc);
  }
  return acc;
}

__device__ __forceinline__ void tile_store(float* __restrict__ D, int sd,
                                           v8f acc, int ti, int tj, int lane) {
  const int rbase = ti * 16 + ((lane >> 4) << 3);
  const int c     = tj * 16 + (lane & 15);
#pragma unroll
  for (int v = 0; v < 8; ++v) D[(rbase + v) * sd + c] = acc[v];
}

// logical element idx (0..4095) -> padded-stride offset
__device__ __forceinline__ int ldof(int idx, int s) {
  return (idx >> 6) * s + (idx & 63);
}

// ---------------------------------------------------------------------------
// K5: CG, single 512-thread workgroup.  ALL operand matrices live in LDS with
// a 76-float padded row stride (bank-conflict-free WMMA operand reads); the
// 8 Kronecker factor matrices are loaded once via async global->LDS DMA.
// Matvec = two WMMA phases (2 barriers); dots use wave32 shuffle reductions.
// ---------------------------------------------------------------------------
__global__ void k_cg(const float* __restrict__ S0g, const float* __restrict__ Qg,
                     const float* __restrict__ Tg,  const float* __restrict__ Urg,
                     const float* __restrict__ Uig, const float* __restrict__ Ng,
                     const float* __restrict__ Prg, const float* __restrict__ Pig,
                     const float* __restrict__ RHS, float* __restrict__ out,
                     int iters) {
  // 0..3: phase-1 B operands {S0, N, Pr, Pi}; 4..7: phase-2 A {Q, T, Ur, Ui}
  __shared__ __attribute__((aligned(16))) float fac[8][K_DIM * LDSTR];
  __shared__ __attribute__((aligned(16))) float Ps [K_DIM * LDSTR];
  __shared__ __attribute__((aligned(16))) float Tm [4][K_DIM * LDSTR];
  __shared__ float Xs [K_DIM * K_DIM];
  __shared__ float Rs [K_DIM * K_DIM];
  __shared__ float Aps[K_DIM * K_DIM];
  __shared__ float red[16];

  const int tid  = threadIdx.x;   // 512
  const int lane = tid & 31;
  const int wave = tid >> 5;
  const int ti   = wave >> 2, tj = wave & 3;

  // One-time async DMA of the 8 factor matrices into padded LDS images.
  {
    const float* gmats[8] = {S0g, Ng, Prg, Pig, Qg, Tg, Urg, Uig};
#pragma unroll
    for (int mm = 0; mm < 8; ++mm) {
#pragma unroll
      for (int h = 0; h < 2; ++h) {
        const int chunk = tid * 2 + h;        // 1024 v4f chunks per matrix
        const int row   = chunk >> 4;
        const int c4    = (chunk & 15) << 2;
        cp16_to_lds(&fac[mm][row * LDSTR + c4],
                    gmats[mm] + row * K_DIM + c4);
      }
    }
  }

#pragma unroll
  for (int i = 0; i < 8; ++i) {
    const int idx = tid + i * 512;
    const float r = RHS[idx];
    Xs[idx] = 0.f;
    Rs[idx] = r;
    Ps[ldof(idx, LDSTR)] = r;
  }
  cp_wait();
  __syncthreads();

  // rr = <r,r>
  float rr;
  {
    float s = 0.f;
#pragma unroll
    for (int i = 0; i < 8; ++i) {
      const int idx = tid + i * 512;
      s += Rs[idx] * Rs[idx];
    }
#pragma unroll
    for (int off = 16; off; off >>= 1) s += __shfl_xor(s, off, 32);
    if (lane == 0) red[wave] = s;
    __syncthreads();
    float tot = 0.f;
#pragma unroll
    for (int w2 = 0; w2 < 16; ++w2) tot += red[w2];
    __syncthreads();
    rr = tot;
  }

  for (int it = 0; it < iters; ++it) {
    // ---- phase 1: Tm[m] = Ps @ fac[m],  m = S0,N,Pr,Pi -----------------
#pragma unroll
    for (int mm = 0; mm < 4; ++mm)
      tile_store(Tm[mm], LDSTR,
                 tile_mm(Ps, LDSTR, fac[mm], LDSTR, ti, tj, lane),
                 ti, tj, lane);
    __syncthreads();
    // ---- phase 2: Ap = Q Tm0 + L(T Tm1 - Ur Tm2 - Ui Tm3) --------------
    const v8f a1 = tile_mm(fac[4], LDSTR, Tm[0], LDSTR, ti, tj, lane);
    const v8f a2 = tile_mm(fac[5], LDSTR, Tm[1], LDSTR, ti, tj, lane);
    const v8f a3 = tile_mm(fac[6], LDSTR, Tm[2], LDSTR, ti, tj, lane);
    const v8f a4 = tile_mm(fac[7], LDSTR, Tm[3], LDSTR, ti, tj, lane);
    {
      const int rbase = ti * 16 + ((lane >> 4) << 3);
      const int c     = tj * 16 + (lane & 15);
#pragma unroll
      for (int v = 0; v < 8; ++v)
        Aps[(rbase + v) * K_DIM + c] = a1[v] + LMBDA * (a2[v] - a3[v] - a4[v]);
    }
    __syncthreads();

    // pAp = <p, Ap>
    float pAp;
    {
      float s = 0.f;
#pragma unroll
      for (int i = 0; i < 8; ++i) {
        const int idx = tid + i * 512;
        s += Ps[ldof(idx, LDSTR)] * Aps[idx];
      }
#pragma unroll
      for (int off = 16; off; off >>= 1) s += __shfl_xor(s, off, 32);
      if (lane == 0) red[wave] = s;
      __syncthreads();
      float tot = 0.f;
#pragma unroll
      for (int w2 = 0; w2 < 16; ++w2) tot += red[w2];
      __syncthreads();
      pAp = tot;
    }
    const float alpha = (pAp != 0.f) ? rr / pAp : 0.f;

    // fused x/r update + ||r||^2 partial (threads touch only own stripes)
    float s = 0.f;
#pragma unroll
    for (int i = 0; i < 8; ++i) {
      const int idx = tid + i * 512;
      Xs[idx] += alpha * Ps[ldof(idx, LDSTR)];
      const float rn = Rs[idx] - alpha * Aps[idx];
      Rs[idx] = rn;
      s += rn * rn;
    }
#pragma unroll
    for (int off = 16; off; off >>= 1) s += __shfl_xor(s, off, 32);
    if (lane == 0) red[wave] = s;
    __syncthreads();
    float rr2 = 0.f;
#pragma unroll
    for (int w2 = 0; w2 < 16; ++w2) rr2 += red[w2];
    __syncthreads();

    const float beta = (rr != 0.f) ? rr2 / rr : 0.f;
    rr = rr2;
#pragma unroll
    for (int i = 0; i < 8; ++i) {
      const int idx = tid + i * 512;
      const int po  = ldof(idx, LDSTR);
      Ps[po] = Rs[idx] + beta * Ps[po];
    }
    __syncthreads();
  }

#pragma unroll
  for (int i = 0; i < 8; ++i) {
    const int idx = tid + i * 512;
    out[idx] = Xs[idx];
  }
}

// ---------------------------------------------------------------------------
extern "C" void kernel_launch(void* const* d_in, const int* in_sizes, int n_in,
                              void* d_out, int out_size, void* d_ws, size_t ws_size,
                              hipStream_t stream) {
  const float* fx = (const float*)d_in[0];
  const float* fy = (const float*)d_in[1];
  const float* ex = (const float*)d_in[2];
  const float* ey = (const float*)d_in[3];
  const float* Px = (const float*)d_in[4];
  const float* Py = (const float*)d_in[5];
  const float* Mx = (const float*)d_in[6];
  const float* My = (const float*)d_in[7];
  float* out = (float*)d_out;

  float* w = (float*)d_ws;
  float* A     = w;              // 64*128
  float* PyF   = w + 8192;       // 64*128
  float* Q     = w + 16384;      // 64*64
  float* AAm   = w + 20480;
  float* RHS   = w + 24576;
  float* invMx = w + 28672;
  float* Nm    = w + 32768;
  float* Pr    = w + 36864;
  float* Pi    = w + 40960;
  float* S0    = w + 45056;
  float* T     = w + 49152;
  float* Ur    = w + 53248;
  float* Ui    = w + 57344;
  float* pbuf  = w + 61440;      // split-K partials: 2*splits*8192 floats

  const size_t need5 = (size_t)(61440 + 2 * SPLITS * 8192) * sizeof(float);
  const int splits = (ws_size >= need5) ? SPLITS : 1;

  k_spectral_gemm<<<dim3(8 * splits), dim3(256), 0, stream>>>(fx, fy, Px, Py,
                                                              pbuf, splits);
  k_reduce_partials<<<dim3(64), dim3(256), 0, stream>>>(pbuf, A, PyF, splits);
  k_smallgemms<<<dim3(1), dim3(256), 0, stream>>>(My, A, PyF, Q, AAm, RHS);
  k_inverse<<<dim3(1), dim3(256), 0, stream>>>(Mx, invMx);
  k_factors<<<dim3(1), dim3(256), 0, stream>>>(ex, ey, invMx, Q, AAm,
                                               Nm, Pr, Pi, S0, T, Ur, Ui);
  k_cg<<<dim3(1), dim3(512), 0, stream>>>(S0, Q, T, Ur, Ui, Nm, Pr, Pi,
                                          RHS, out, CG_ITERS);
}